// ConvLSTM_75960791597163
// MI455X (gfx1250) — compile-verified
//
#include <hip/hip_runtime.h>
#include <math.h>

// ---------------- problem constants (from reference) ----------------
#define HIDDEN   128
#define CIN      64
#define CCOMB    (CIN + HIDDEN)        // 192
#define KTAPS    5
#define KDIM     (CCOMB * KTAPS)       // 960
#define BATCH    16
#define TSTEPS   64
#define WDIM     256
#define NCOLS    (BATCH * WDIM)        // 4096 GEMM columns per step

// ---------------- tiling ----------------
#define KT       64                    // K-chunk staged through LDS
#define NCHUNK   (KDIM / KT)           // 15
#define NT       64                    // N tile per workgroup
#define HCT      32                    // hidden-channel tile (x4 gates = 128 M rows)
#define A_PAD    68                    // A row stride in LDS: 4l / 4l+2 bank pattern, conflict-free frag reads
#define BP_STR   160                   // K-pair row stride: 160 % 64 == 32 -> half-waves on disjoint banks

typedef __attribute__((ext_vector_type(2))) float v2f;
typedef __attribute__((ext_vector_type(4))) float v4f;
typedef __attribute__((ext_vector_type(8))) float v8f;

__device__ __forceinline__ float fast_sigmoid(float x) {
    return 1.0f / (1.0f + __expf(-x));
}
__device__ __forceinline__ float fast_tanh(float x) {
    return 2.0f / (1.0f + __expf(-2.0f * x)) - 1.0f;
}

// One ConvLSTM timestep as implicit GEMM: D[o,n] = sum_k A[o,k]*Bmat[k,n], o=gate*128+hc,
// n=b*256+w, k=cin*5+tap; Bmat[k,n] = concat(x_t, h_{t-1})[b, cin, w+tap-2] (0-padded).
__global__ __launch_bounds__(256)
void convlstm_step(const float* __restrict__ x,       // [B, T, CIN, W]
                   const float* __restrict__ Wc,      // [512, 192, 5] row-major
                   const float* __restrict__ bc,      // [512]
                   float* __restrict__ out,           // [B, T, H, W]
                   float* __restrict__ c_ws,          // [B, H, W]
                   float* __restrict__ h_final,       // [B, H, W]
                   float* __restrict__ c_final,       // [B, H, W]
                   int t)
{
    __shared__ float sA[4 * HCT * A_PAD];      // 128 x 64 (padded)  ~34.8 KB
    __shared__ float sB[(KT / 2) * BP_STR];    // 32 K-pairs x 64 cols interleaved  20 KB

    const int tid  = threadIdx.x;
    const int lane = tid & 31;
    const int wave = tid >> 5;
    const int gate = wave & 3;                 // LSTM gate handled by this wave
    const int sub  = wave >> 2;                // 16-row half of the 32-hc tile
    const int hi   = lane >> 4;                // WMMA half-wave select
    const int lm   = lane & 15;

    const int n0  = blockIdx.x * NT;
    const int hc0 = blockIdx.y * HCT;

    // ---- per-thread invariant staging coordinates ----
    // A: thread stores 8 x float4; column block fixed, row varies with i
    const int a_kkc  = (tid & 15) << 2;        // 0,4,...,60
    const int a_row0 = tid >> 4;               // + 16*i
    // B: thread stores 8 x K-pair (v2f); column fixed, pair row varies with i
    const int b_n    = tid & 63;
    const int b_p0   = tid >> 6;               // + 4*i
    const int b_ng   = n0 + b_n;
    const int b_b    = b_ng >> 8;              // batch of this column
    const int b_w    = b_ng & 255;             // position of this column

    const float* xb = x + ((size_t)b_b * TSTEPS + t) * CIN * WDIM;          // x[b, t, :, :]
    const float* hb = (t > 0)
        ? out + ((size_t)b_b * TSTEPS + (t - 1)) * HIDDEN * WDIM            // h_{t-1}
        : nullptr;

    v4f aReg[8];
    v2f bReg[8];

    auto loadA = [&](int k0) {
        #pragma unroll
        for (int i = 0; i < 8; ++i) {
            int row = a_row0 + 16 * i;                 // 0..127
            int o   = (row >> 5) * HIDDEN + hc0 + (row & 31);
            aReg[i] = *(const v4f*)&Wc[(size_t)o * KDIM + k0 + a_kkc];
        }
    };
    auto loadB = [&](int k0) {
        #pragma unroll
        for (int i = 0; i < 8; ++i) {
            int p = b_p0 + 4 * i;                      // K-pair row 0..31
            float e[2];
            #pragma unroll
            for (int eidx = 0; eidx < 2; ++eidx) {
                int kabs = k0 + 2 * p + eidx;
                int cin  = kabs / KTAPS;
                int tap  = kabs - cin * KTAPS;
                int w    = b_w + tap - 2;              // 'same' padding, K=5
                float v  = 0.0f;
                if (w >= 0 && w < WDIM) {
                    if (cin < CIN)       v = xb[cin * WDIM + w];
                    else if (hb)         v = hb[(cin - CIN) * WDIM + w];
                }
                e[eidx] = v;
            }
            bReg[i].x = e[0];
            bReg[i].y = e[1];
        }
    };
    auto storeLDS = [&]() {
        #pragma unroll
        for (int i = 0; i < 8; ++i) {
            int row = a_row0 + 16 * i;
            *(v4f*)&sA[row * A_PAD + a_kkc] = aReg[i];
        }
        #pragma unroll
        for (int i = 0; i < 8; ++i) {
            int p = b_p0 + 4 * i;
            *(v2f*)&sB[p * BP_STR + 2 * b_n] = bReg[i];
        }
    };

    v8f acc0 = {}, acc1 = {}, acc2 = {}, acc3 = {};

    // fragment base pointers (A includes half-wave K offset; B column offset in pair units)
    const float* aRow = &sA[(gate * HCT + sub * 16 + lm) * A_PAD + 2 * hi];
    const float* bCol = &sB[hi * BP_STR + 2 * lm];

    loadA(0);
    loadB(0);

    for (int chunk = 0; chunk < NCHUNK; ++chunk) {
        storeLDS();
        __syncthreads();
        if (chunk + 1 < NCHUNK) {                  // prefetch next chunk during WMMA
            loadA((chunk + 1) * KT);
            loadB((chunk + 1) * KT);
        }
        #pragma unroll
        for (int kb = 0; kb < KT / 4; ++kb) {
            // A frag (16x4): lanes 0-15 K={kc,kc+1}, lanes 16-31 K={kc+2,kc+3}
            v2f a = *(const v2f*)&aRow[kb * 4];
            // B frags (4x16): one ds_load_b64 each thanks to K-pair interleave
            const float* bp = &bCol[(kb * 2) * BP_STR];
            v2f bf0 = *(const v2f*)&bp[0];
            v2f bf1 = *(const v2f*)&bp[32];
            v2f bf2 = *(const v2f*)&bp[64];
            v2f bf3 = *(const v2f*)&bp[96];
            acc0 = __builtin_amdgcn_wmma_f32_16x16x4_f32(false, a, false, bf0, (short)0, acc0, false, false);
            acc1 = __builtin_amdgcn_wmma_f32_16x16x4_f32(false, a, false, bf1, (short)0, acc1, false, false);
            acc2 = __builtin_amdgcn_wmma_f32_16x16x4_f32(false, a, false, bf2, (short)0, acc2, false, false);
            acc3 = __builtin_amdgcn_wmma_f32_16x16x4_f32(false, a, false, bf3, (short)0, acc3, false, false);
        }
        __syncthreads();
    }

    // ---- epilogue: spill accumulators to LDS (reuse A region), fuse LSTM gates ----
    // D layout: lane l, VGPR v -> row M = v + 8*(l>>4), col N = l&15
    float* cc = sA;  // [4 gates][32 hc][64 n], stride A_PAD
    {
        const int rbase = gate * HCT + sub * 16 + 8 * hi;
        #pragma unroll
        for (int v = 0; v < 8; ++v) {
            cc[(rbase + v) * A_PAD + 0 * 16 + lm] = acc0[v];
            cc[(rbase + v) * A_PAD + 1 * 16 + lm] = acc1[v];
            cc[(rbase + v) * A_PAD + 2 * 16 + lm] = acc2[v];
            cc[(rbase + v) * A_PAD + 3 * 16 + lm] = acc3[v];
        }
    }
    __syncthreads();

    const bool last = (t == TSTEPS - 1);
    #pragma unroll
    for (int i = 0; i < (HCT * NT) / 256; ++i) {
        int idx = tid + i * 256;                 // 0 .. 2047
        int hcl = idx >> 6;
        int n   = idx & 63;
        int hc  = hc0 + hcl;
        float ci = cc[(0 * HCT + hcl) * A_PAD + n] + bc[0 * HIDDEN + hc];
        float cf = cc[(1 * HCT + hcl) * A_PAD + n] + bc[1 * HIDDEN + hc];
        float co = cc[(2 * HCT + hcl) * A_PAD + n] + bc[2 * HIDDEN + hc];
        float cg = cc[(3 * HCT + hcl) * A_PAD + n] + bc[3 * HIDDEN + hc];

        int ng = n0 + n;
        int b  = ng >> 8;
        int w  = ng & 255;
        size_t sidx = ((size_t)b * HIDDEN + hc) * WDIM + w;

        float cprev = (t == 0) ? 0.0f : c_ws[sidx];
        float iv = fast_sigmoid(ci);
        float fv = fast_sigmoid(cf);
        float ov = fast_sigmoid(co);
        float gv = fast_tanh(cg);
        float cn = fv * cprev + iv * gv;
        float hn = ov * fast_tanh(cn);

        c_ws[sidx] = cn;
        out[(((size_t)b * TSTEPS + t) * HIDDEN + hc) * WDIM + w] = hn;
        if (last) {
            h_final[sidx] = hn;
            c_final[sidx] = cn;
        }
    }
}

extern "C" void kernel_launch(void* const* d_in, const int* in_sizes, int n_in,
                              void* d_out, int out_size, void* d_ws, size_t ws_size,
                              hipStream_t stream) {
    (void)in_sizes; (void)n_in; (void)out_size; (void)ws_size;
    const float* x  = (const float*)d_in[0];   // [16, 64, 64, 256]
    const float* Wc = (const float*)d_in[1];   // [512, 192, 5]
    const float* bc = (const float*)d_in[2];   // [512]
    float* out = (float*)d_out;                // outputs | h | c, concatenated flat
    float* h_final = out + (size_t)BATCH * TSTEPS * HIDDEN * WDIM;
    float* c_final = h_final + (size_t)BATCH * HIDDEN * WDIM;
    float* c_ws = (float*)d_ws;                // [16, 128, 256] cell state

    dim3 grid(NCOLS / NT, HIDDEN / HCT);       // (64, 4) workgroups per step
    for (int t = 0; t < TSTEPS; ++t) {
        convlstm_step<<<grid, 256, 0, stream>>>(x, Wc, bc, out, c_ws, h_final, c_final, t);
    }
}